// VEG_87729001988776
// MI455X (gfx1250) — compile-verified
//
#include <hip/hip_runtime.h>
#include <hip/hip_bf16.h>

#define N_PTS   262144
#define N_GAUSS 2048
#define NZ 100
#define NY 100
#define NX 100
#define MAX_SCALE 0.02f
#define BG (-1.0f)

// Fold exp(-0.5*qd) = exp2(C*qd) with C = -0.5*log2(e) into the coefficients.
#define NEG_HALF_LOG2E (-0.72134752044448170368f)
// qd <= 9  <=>  C*qd >= 9*C   (C < 0 flips the inequality)
#define LOG2_CUTOFF (-6.4921276840003353331f)

typedef float v2f __attribute__((ext_vector_type(2)));
typedef float v8f __attribute__((ext_vector_type(8)));

__device__ __forceinline__ float sigmoidf_(float x) {
    return 1.0f / (1.0f + __expf(-x));
}

// ---------------------------------------------------------------------------
// Kernel 1: per-Gaussian coefficient precompute.
// coef[g][12] = C * {A00, A11, A22, 2A01, 2A02, 2A12, -2b0, -2b1, -2b2, c}
//               ++ {w, wv}            (slots 10,11 unscaled; feature = 0 there)
// where A = R diag(1/s^2) R^T, b = A*mu, c = mu^T A mu, C = -0.5*log2(e).
// ---------------------------------------------------------------------------
__global__ __launch_bounds__(256) void veg_coef_kernel(
    const float* __restrict__ xyz,
    const float* __restrict__ scaling_raw,
    const float* __restrict__ rotation_raw,
    const float* __restrict__ weight_raw,
    const float* __restrict__ values_raw,
    float* __restrict__ coef)
{
    int g = blockIdx.x * 256 + threadIdx.x;
    if (g >= N_GAUSS) return;

    // scales with soft radius clamp
    float s0 = __expf(scaling_raw[3*g+0]);
    float s1 = __expf(scaling_raw[3*g+1]);
    float s2 = __expf(scaling_raw[3*g+2]);
    float r  = sqrtf(s0*s0 + s1*s1 + s2*s2) + 1e-8f;
    float rsoft = MAX_SCALE * tanhf(r / MAX_SCALE);
    float sc = rsoft / r;
    s0 *= sc; s1 *= sc; s2 *= sc;

    // normalized quaternion (r,x,y,z)
    float qr = rotation_raw[4*g+0];
    float qx = rotation_raw[4*g+1];
    float qy = rotation_raw[4*g+2];
    float qz = rotation_raw[4*g+3];
    float qn = fmaxf(sqrtf(qr*qr + qx*qx + qy*qy + qz*qz), 1e-12f);
    float inv = 1.0f / qn;
    qr *= inv; qx *= inv; qy *= inv; qz *= inv;

    float R00 = 1.0f - 2.0f*(qy*qy + qz*qz);
    float R01 = 2.0f*(qx*qy - qr*qz);
    float R02 = 2.0f*(qx*qz + qr*qy);
    float R10 = 2.0f*(qx*qy + qr*qz);
    float R11 = 1.0f - 2.0f*(qx*qx + qz*qz);
    float R12 = 2.0f*(qy*qz - qr*qx);
    float R20 = 2.0f*(qx*qz - qr*qy);
    float R21 = 2.0f*(qy*qz + qr*qx);
    float R22 = 1.0f - 2.0f*(qx*qx + qy*qy);

    float d0 = 1.0f/(s0*s0), d1 = 1.0f/(s1*s1), d2 = 1.0f/(s2*s2);

    // A = R diag(d) R^T  (symmetric)
    float A00 = R00*d0*R00 + R01*d1*R01 + R02*d2*R02;
    float A11 = R10*d0*R10 + R11*d1*R11 + R12*d2*R12;
    float A22 = R20*d0*R20 + R21*d1*R21 + R22*d2*R22;
    float A01 = R00*d0*R10 + R01*d1*R11 + R02*d2*R12;
    float A02 = R00*d0*R20 + R01*d1*R21 + R02*d2*R22;
    float A12 = R10*d0*R20 + R11*d1*R21 + R12*d2*R22;

    float m0 = xyz[3*g+0], m1 = xyz[3*g+1], m2 = xyz[3*g+2];
    float b0 = A00*m0 + A01*m1 + A02*m2;
    float b1 = A01*m0 + A11*m1 + A12*m2;
    float b2 = A02*m0 + A12*m1 + A22*m2;
    float c  = m0*b0 + m1*b1 + m2*b2;

    float w  = sigmoidf_(weight_raw[g]);
    float v  = sigmoidf_(values_raw[g]);

    const float C = NEG_HALF_LOG2E;   // fold exp scaling into the GEMM
    float* row = coef + g * 12;
    row[0]  = C*A00;        row[1]  = C*A11;        row[2]  = C*A22;
    row[3]  = C*2.0f*A01;   row[4]  = C*2.0f*A02;   row[5]  = C*2.0f*A12;
    row[6]  = C*-2.0f*b0;   row[7]  = C*-2.0f*b1;   row[8]  = C*-2.0f*b2;
    row[9]  = C*c;          row[10] = w;            row[11] = w*v;
}

// ---------------------------------------------------------------------------
// Kernel 2: main WMMA kernel. Each wave owns 16 points x all 2048 Gaussians.
// log2(g)(point, gauss) via 3x V_WMMA_F32_16X16X4_F32 (K=12, features padded),
// then per-pair: exp2 + cutoff select + 2 FMAs.
// ---------------------------------------------------------------------------
__global__ __launch_bounds__(256) void veg_main_kernel(
    const float* __restrict__ x,
    const float* __restrict__ coef,
    const unsigned char* __restrict__ valid_grid,
    float* __restrict__ out)
{
    const int tid  = threadIdx.x;
    const int wave = tid >> 5;
    const int lane = tid & 31;
    const int l    = lane & 15;         // column within half
    const int half = lane >> 4;         // which K pair of each K=4 tile
    const int pt0  = blockIdx.x * 128 + wave * 16;

    // ---- A fragments: features of point pt0+l, split across K halves ----
    const float* xp = x + 3 * (pt0 + l);
    float X = (xp[0] + 1.0f) * 0.5f;
    float Y = (xp[1] + 1.0f) * 0.5f;
    float Z = (xp[2] + 1.0f) * 0.5f;

    const bool hi = (half != 0);
    v2f a0, a1, a2;
    // K layout per lane: {2*half, 2*half+1} within each K=4 tile.
    // f = [x2, y2, z2, xy, xz, yz, x, y, z, 1, 0, 0]
    a0.x = hi ? Z*Z  : X*X;   a0.y = hi ? X*Y  : Y*Y;   // K tile 0: f0..f3
    a1.x = hi ? X    : X*Z;   a1.y = hi ? Y    : Y*Z;   // K tile 1: f4..f7
    a2.x = hi ? 0.0f : Z;     a2.y = hi ? 0.0f : 1.0f;  // K tile 2: f8..f11

    float den[8], num[8];
#pragma unroll
    for (int j = 0; j < 8; ++j) { den[j] = 0.0f; num[j] = 0.0f; }

    const int ko = half * 2;

    for (int t = 0; t < N_GAUSS / 16; ++t) {
        const float* crow = coef + (t * 16 + l) * 12;
        v2f b0  = *(const v2f*)(crow + 0 + ko);
        v2f b1  = *(const v2f*)(crow + 4 + ko);
        v2f b2  = *(const v2f*)(crow + 8 + ko);   // half=1 reads (w,wv) but a2==0
        v2f wwv = *(const v2f*)(crow + 10);       // (w, w*v) for this gaussian column

        v8f c = {};
        c = __builtin_amdgcn_wmma_f32_16x16x4_f32(false, a0, false, b0, (short)0, c, false, false);
        c = __builtin_amdgcn_wmma_f32_16x16x4_f32(false, a1, false, b1, (short)0, c, false, false);
        c = __builtin_amdgcn_wmma_f32_16x16x4_f32(false, a2, false, b2, (short)0, c, false, false);

#pragma unroll
        for (int j = 0; j < 8; ++j) {
            float q  = c[j];                         // already -0.5*log2(e)*qd
            float e  = __builtin_amdgcn_exp2f(q);
            float gq = (q >= LOG2_CUTOFF) ? e : 0.0f;
            den[j] = fmaf(gq, wwv.x, den[j]);
            num[j] = fmaf(gq, wwv.y, num[j]);
        }
    }

    // ---- reduce over the 16 gaussian columns in each half, then store ----
#pragma unroll
    for (int j = 0; j < 8; ++j) {
        float dn = den[j];
        float nm = num[j];
#pragma unroll
        for (int m = 8; m >= 1; m >>= 1) {
            dn += __shfl_xor(dn, m, 32);   // masks <= 8 never cross the 16-lane halves
            nm += __shfl_xor(nm, m, 32);
        }
        if (l == j) {
            int p = pt0 + j + 8 * half;    // C-layout row: M = vgpr + 8*(lane/16)
            const float* xq = x + 3 * p;
            float px = (xq[0] + 1.0f) * 0.5f;
            float py = (xq[1] + 1.0f) * 0.5f;
            float pz = (xq[2] + 1.0f) * 0.5f;
            int ix = min(max((int)rintf(px * (NX - 1)), 0), NX - 1);
            int iy = min(max((int)rintf(py * (NY - 1)), 0), NY - 1);
            int iz = min(max((int)rintf(pz * (NZ - 1)), 0), NZ - 1);
            int flat = iz * (NY * NX) + iy * NX + ix;
            bool valid = valid_grid[flat] != 0;
            float y = (dn > 0.0f) ? (nm / dn) : BG;
            out[p] = valid ? y : 0.0f;
        }
    }
}

extern "C" void kernel_launch(void* const* d_in, const int* in_sizes, int n_in,
                              void* d_out, int out_size, void* d_ws, size_t ws_size,
                              hipStream_t stream) {
    const float* x            = (const float*)d_in[0];
    const float* xyz          = (const float*)d_in[1];
    const float* scaling_raw  = (const float*)d_in[2];
    const float* rotation_raw = (const float*)d_in[3];
    const float* weight_raw   = (const float*)d_in[4];
    const float* values_raw   = (const float*)d_in[5];
    const unsigned char* valid_grid = (const unsigned char*)d_in[6];
    float* out  = (float*)d_out;
    float* coef = (float*)d_ws;   // needs N_GAUSS*12*4 = 96 KB of scratch

    veg_coef_kernel<<<(N_GAUSS + 255) / 256, 256, 0, stream>>>(
        xyz, scaling_raw, rotation_raw, weight_raw, values_raw, coef);

    veg_main_kernel<<<N_PTS / 128, 256, 0, stream>>>(x, coef, valid_grid, out);
}